// EGIN_56599079027147
// MI455X (gfx1250) — compile-verified
//
#include <hip/hip_runtime.h>

// EGIN forward for MI455X (gfx1250, wave32).
// Memory-bound problem (~6-7 GB traffic -> ~300us floor @ 23.3 TB/s); GEMMs are
// only ~39 GFLOP, so we use full-precision V_WMMA_F32_16X16X4_F32 (exact f32)
// rather than low-precision matrix ops -- compute is free here, accuracy isn't.
// GEMM A-tiles are staged in LDS (coalesced b128 global loads, padded stride for
// bank-conflict-free ds_load_b64 fragment reads); BN+relu for GEMM2's A operand
// is applied once per element at staging time.

typedef __attribute__((ext_vector_type(2))) float v2f;
typedef __attribute__((ext_vector_type(4))) float v4f;
typedef __attribute__((ext_vector_type(8))) float v8f;

static constexpr int NNODES = 100000;
static constexpr int NEDGES = 1600000;
static constexpr int D      = 128;
static constexpr int D2     = 256;
static constexpr int NG     = 1024;
static constexpr int ATOMV  = 100;
static constexpr int BONDV  = 6;

// LDS row strides (floats), padded so stride % 64 == 4 banks:
// fragment read banks = kb + 4*row (+2*half) mod 64 -> 64 distinct dwords.
static constexpr int SA1 = D  + 4;   // 132
static constexpr int SA2 = D2 + 4;   // 260

// ---------------------------------------------------------------- atom encoder
__global__ __launch_bounds__(256) void k_atom_enc(const int* __restrict__ x,
                                                  const float* __restrict__ aemb,
                                                  float* __restrict__ h) {
  int tid  = blockIdx.x * 256 + threadIdx.x;
  int n    = tid >> 5;
  int lane = tid & 31;
  const int* xr = x + n * 9;
  v4f acc = {0.f, 0.f, 0.f, 0.f};
#pragma unroll
  for (int f = 0; f < 9; ++f) {
    int idx = xr[f];
    acc += ((const v4f*)(aemb + (f * ATOMV + idx) * D))[lane];
  }
  ((v4f*)(h + n * D))[lane] = acc;
}

// ---------------------------------------------------------------- edge message
__global__ __launch_bounds__(256) void k_edge(const int* __restrict__ ex,
                                              const int* __restrict__ src,
                                              const int* __restrict__ dst,
                                              const float* __restrict__ h,
                                              const float* __restrict__ bemb,
                                              float* __restrict__ agg) {
  int tid  = blockIdx.x * 256 + threadIdx.x;
  int e    = tid >> 5;
  int lane = tid & 31;
  int s = src[e];
  int d = dst[e];
  const int* exr = ex + e * 3;
  v4f m = ((const v4f*)(h + (size_t)s * D))[lane];
#pragma unroll
  for (int f = 0; f < 3; ++f) {
    m += ((const v4f*)(bemb + (f * BONDV + exr[f]) * D))[lane];
  }
  float* out = agg + (size_t)d * D + lane * 4;
#pragma unroll
  for (int j = 0; j < 4; ++j) {
    atomicAdd(out + j, fmaxf(m[j], 0.0f));
  }
}

// ---------------------------------------------------------------- z = (1+eps)*h + agg  (in place into agg)
__global__ __launch_bounds__(256) void k_zcomb(const float* __restrict__ h,
                                               float* __restrict__ agg,
                                               const float* __restrict__ eps,
                                               int layer) {
  int idx = blockIdx.x * 256 + threadIdx.x;
  float s = 1.0f + eps[layer];
  agg[idx] = fmaf(s, h[idx], agg[idx]);
}

// ---------------------------------------------------------------- GEMM1: t = z @ W1 + b1, fused BN stats
// 512 threads = 16 waves = all 16 n-tiles of one 16-row m-tile.
// A tile (16x128) staged once in LDS, WMMA fragments fed from ds_load_b64.
__global__ __launch_bounds__(512) void k_gemm1(const float* __restrict__ z,
                                               const float* __restrict__ W1,
                                               const float* __restrict__ b1,
                                               float* __restrict__ t,
                                               float* __restrict__ stats) {
  __shared__ float sA[16 * SA1];
  int mtile = blockIdx.x;

  // stage: 2048 floats / 512 threads = one float4 each (coalesced b128 loads)
  {
    int fi  = threadIdx.x;        // float4 slot
    int row = fi >> 5;            // 32 float4 per row
    int cv  = (fi & 31) * 4;
    v4f val = *(const v4f*)(z + (size_t)(mtile * 16 + row) * D + cv);
    v2f lo = {val.x, val.y};
    v2f hi = {val.z, val.w};
    *(v2f*)&sA[row * SA1 + cv]     = lo;
    *(v2f*)&sA[row * SA1 + cv + 2] = hi;
  }
  __syncthreads();

  int lane  = threadIdx.x & 31;
  int half  = lane >> 4;
  int l16   = lane & 15;
  int ntile = threadIdx.x >> 5;   // 0..15
  int col   = ntile * 16 + l16;

  v8f c;
  float bv = b1[col];
#pragma unroll
  for (int v = 0; v < 8; ++v) c[v] = bv;

  const float* arow = &sA[l16 * SA1];
#pragma unroll 4
  for (int k = 0; k < D; k += 4) {
    int kb = k + half * 2;
    v2f a = *(const v2f*)(arow + kb);
    v2f b;
    b.x = W1[kb * D2 + col];
    b.y = W1[(kb + 1) * D2 + col];
    c = __builtin_amdgcn_wmma_f32_16x16x4_f32(false, a, false, b, (short)0, c,
                                              false, false);
  }

  int rbase = mtile * 16 + half * 8;
  float s = 0.f, sq = 0.f;
#pragma unroll
  for (int v = 0; v < 8; ++v) {
    float val = c[v];
    t[(size_t)(rbase + v) * D2 + col] = val;
    s  += val;
    sq += val * val;
  }
  s  += __shfl_xor(s, 16, 32);
  sq += __shfl_xor(sq, 16, 32);
  if (half == 0) {
    atomicAdd(&stats[col], s);
    atomicAdd(&stats[D2 + col], sq);
  }
}

// ---------------------------------------------------------------- BN finalize
__global__ void k_bnfin(const float* __restrict__ stats,
                        const float* __restrict__ g,
                        const float* __restrict__ be,
                        float* __restrict__ scale,
                        float* __restrict__ shift,
                        int C, float invN) {
  int cI = threadIdx.x;
  float mu  = stats[cI] * invN;
  float var = stats[C + cI] * invN - mu * mu;
  float rs  = rsqrtf(var + 1e-5f);
  float sc  = g[cI] * rs;
  scale[cI] = sc;
  shift[cI] = be[cI] - mu * sc;
}

// ---------------------------------------------------------------- GEMM2: h = relu(bn(t)) @ W2 + b2, fused stats
// 256 threads = 8 waves = 8 n-tiles. BN+relu applied once per element while
// staging the 16x256 A tile into LDS.
__global__ __launch_bounds__(256) void k_gemm2(const float* __restrict__ t,
                                               const float* __restrict__ scale,
                                               const float* __restrict__ shift,
                                               const float* __restrict__ W2,
                                               const float* __restrict__ b2,
                                               float* __restrict__ hout,
                                               float* __restrict__ stats2,
                                               int do_stats) {
  __shared__ float sA[16 * SA2];
  int mtile = blockIdx.x;

  // stage: 4096 floats / 256 threads = 4 float4 each, act() fused here
#pragma unroll
  for (int r = 0; r < 4; ++r) {
    int fi  = threadIdx.x + r * 256;  // float4 slot
    int row = fi >> 6;                // 64 float4 per row
    int k0  = (fi & 63) * 4;
    v4f tv = *(const v4f*)(t + (size_t)(mtile * 16 + row) * D2 + k0);
    v2f lo, hi;
    lo.x = fmaxf(fmaf(scale[k0],     tv.x, shift[k0]),     0.f);
    lo.y = fmaxf(fmaf(scale[k0 + 1], tv.y, shift[k0 + 1]), 0.f);
    hi.x = fmaxf(fmaf(scale[k0 + 2], tv.z, shift[k0 + 2]), 0.f);
    hi.y = fmaxf(fmaf(scale[k0 + 3], tv.w, shift[k0 + 3]), 0.f);
    *(v2f*)&sA[row * SA2 + k0]     = lo;
    *(v2f*)&sA[row * SA2 + k0 + 2] = hi;
  }
  __syncthreads();

  int lane  = threadIdx.x & 31;
  int half  = lane >> 4;
  int l16   = lane & 15;
  int ntile = threadIdx.x >> 5;   // 0..7
  int col   = ntile * 16 + l16;

  v8f c;
  float bv = b2[col];
#pragma unroll
  for (int v = 0; v < 8; ++v) c[v] = bv;

  const float* arow = &sA[l16 * SA2];
#pragma unroll 4
  for (int k = 0; k < D2; k += 4) {
    int kb = k + half * 2;
    v2f a = *(const v2f*)(arow + kb);
    v2f b;
    b.x = W2[kb * D + col];
    b.y = W2[(kb + 1) * D + col];
    c = __builtin_amdgcn_wmma_f32_16x16x4_f32(false, a, false, b, (short)0, c,
                                              false, false);
  }

  int rbase = mtile * 16 + half * 8;
  float s = 0.f, sq = 0.f;
#pragma unroll
  for (int v = 0; v < 8; ++v) {
    float val = c[v];
    hout[(size_t)(rbase + v) * D + col] = val;
    s  += val;
    sq += val * val;
  }
  s  += __shfl_xor(s, 16, 32);
  sq += __shfl_xor(sq, 16, 32);
  if (do_stats && half == 0) {
    atomicAdd(&stats2[col], s);
    atomicAdd(&stats2[D + col], sq);
  }
}

// ---------------------------------------------------------------- apply BN2 + relu in place
__global__ __launch_bounds__(256) void k_bnapply(float* __restrict__ h,
                                                 const float* __restrict__ sc,
                                                 const float* __restrict__ sh) {
  int idx = blockIdx.x * 256 + threadIdx.x;
  int cI  = idx & (D - 1);
  h[idx] = fmaxf(fmaf(sc[cI], h[idx], sh[cI]), 0.f);
}

// ---------------------------------------------------------------- pooling scatter
__global__ __launch_bounds__(256) void k_pool(const float* __restrict__ h,
                                              const int* __restrict__ ng,
                                              float* __restrict__ pooled,
                                              float* __restrict__ cnt) {
  int tid  = blockIdx.x * 256 + threadIdx.x;
  int n    = tid >> 5;
  int lane = tid & 31;
  int g = ng[n];
  v4f hv = ((const v4f*)(h + (size_t)n * D))[lane];
  float* out = pooled + g * D + lane * 4;
#pragma unroll
  for (int j = 0; j < 4; ++j) atomicAdd(out + j, hv[j]);
  if (lane == 0) atomicAdd(&cnt[g], 1.0f);
}

// ---------------------------------------------------------------- output GEMM: (pooled/cnt) @ outW + outb
__global__ __launch_bounds__(256) void k_outgemm(const float* __restrict__ pooled,
                                                 const float* __restrict__ cnt,
                                                 const float* __restrict__ outW,
                                                 const float* __restrict__ outb,
                                                 float* __restrict__ out) {
  int lane  = threadIdx.x & 31;
  int half  = lane >> 4;
  int l16   = lane & 15;
  int ntile = threadIdx.x >> 5;
  int mtile = blockIdx.x;
  int col   = ntile * 16 + l16;
  int m     = mtile * 16 + l16;   // A-fragment row for this lane
  float inv = 1.0f / fmaxf(cnt[m], 1.0f);

  v8f c;
  float bv = outb[col];
#pragma unroll
  for (int v = 0; v < 8; ++v) c[v] = bv;

  const float* prow = pooled + (size_t)m * D;
#pragma unroll 4
  for (int k = 0; k < D; k += 4) {
    int kb = k + half * 2;
    v2f pv = *(const v2f*)(prow + kb);
    v2f a = {pv.x * inv, pv.y * inv};
    v2f b = {outW[kb * D + col], outW[(kb + 1) * D + col]};
    c = __builtin_amdgcn_wmma_f32_16x16x4_f32(false, a, false, b, (short)0, c,
                                              false, false);
  }

  int rbase = mtile * 16 + half * 8;
#pragma unroll
  for (int v = 0; v < 8; ++v) {
    out[(size_t)(rbase + v) * D + col] = c[v];
  }
}

// ================================================================ host
extern "C" void kernel_launch(void* const* d_in, const int* in_sizes, int n_in,
                              void* d_out, int out_size, void* d_ws, size_t ws_size,
                              hipStream_t stream) {
  (void)in_sizes; (void)n_in; (void)out_size; (void)ws_size;

  const int*   x        = (const int*)d_in[0];
  const int*   ex       = (const int*)d_in[1];
  const int*   src      = (const int*)d_in[2];
  const int*   dst      = (const int*)d_in[3];
  const int*   ng       = (const int*)d_in[4];
  const float* atom_emb = (const float*)d_in[5];
  const float* bond_emb = (const float*)d_in[6];
  const float* eps      = (const float*)d_in[7];
  const float* W1       = (const float*)d_in[8];
  const float* b1       = (const float*)d_in[9];
  const float* g1       = (const float*)d_in[10];
  const float* be1      = (const float*)d_in[11];
  const float* W2       = (const float*)d_in[12];
  const float* b2       = (const float*)d_in[13];
  const float* bn_g     = (const float*)d_in[14];
  const float* bn_b     = (const float*)d_in[15];
  const float* outW     = (const float*)d_in[16];
  const float* outb     = (const float*)d_in[17];
  float* out = (float*)d_out;

  // workspace carving (floats)
  float* ws     = (float*)d_ws;
  float* h      = ws;                              // N*D
  float* agg    = h      + (size_t)NNODES * D;     // N*D  (also holds z)
  float* t      = agg    + (size_t)NNODES * D;     // N*2D
  float* stats  = t      + (size_t)NNODES * D2;    // 2*256
  float* scale  = stats  + 2 * D2;                 // 256
  float* shift  = scale  + D2;                     // 256
  float* stats2 = shift  + D2;                     // 2*128
  float* scale2 = stats2 + 2 * D;                  // 128
  float* shift2 = scale2 + D;                      // 128
  float* pooled = shift2 + D;                      // G*D
  float* cnt    = pooled + (size_t)NG * D;         // G

  const int nodeWaveBlocks = NNODES * 32 / 256;    // 12500
  const int edgeWaveBlocks = NEDGES * 32 / 256;    // 200000
  const int elemBlocks     = NNODES * D / 256;     // 50000
  const int mTiles         = NNODES / 16;          // 6250

  k_atom_enc<<<nodeWaveBlocks, 256, 0, stream>>>(x, atom_emb, h);

  for (int i = 0; i < 3; ++i) {
    hipMemsetAsync(agg,    0, (size_t)NNODES * D * sizeof(float), stream);
    hipMemsetAsync(stats,  0, 2 * D2 * sizeof(float), stream);
    hipMemsetAsync(stats2, 0, 2 * D * sizeof(float), stream);

    k_edge<<<edgeWaveBlocks, 256, 0, stream>>>(
        ex, src, dst, h, bond_emb + (size_t)i * 3 * BONDV * D, agg);
    k_zcomb<<<elemBlocks, 256, 0, stream>>>(h, agg, eps, i);
    k_gemm1<<<mTiles, 512, 0, stream>>>(
        agg, W1 + (size_t)i * D * D2, b1 + (size_t)i * D2, t, stats);
    k_bnfin<<<1, D2, 0, stream>>>(stats, g1 + (size_t)i * D2, be1 + (size_t)i * D2,
                                  scale, shift, D2, 1.0f / (float)NNODES);
    k_gemm2<<<mTiles, 256, 0, stream>>>(
        t, scale, shift, W2 + (size_t)i * D2 * D, b2 + (size_t)i * D, h, stats2,
        (i < 2) ? 1 : 0);
    if (i < 2) {
      k_bnfin<<<1, D, 0, stream>>>(stats2, bn_g + (size_t)i * D, bn_b + (size_t)i * D,
                                   scale2, shift2, D, 1.0f / (float)NNODES);
      k_bnapply<<<elemBlocks, 256, 0, stream>>>(h, scale2, shift2);
    }
  }

  hipMemsetAsync(pooled, 0, ((size_t)NG * D + NG) * sizeof(float), stream);
  k_pool<<<nodeWaveBlocks, 256, 0, stream>>>(h, ng, pooled, cnt);
  k_outgemm<<<NG / 16, 256, 0, stream>>>(pooled, cnt, outW, outb, out);
}